// MyModel_87522843561267
// MI455X (gfx1250) — compile-verified
//
#include <hip/hip_runtime.h>
#include <math.h>

// Problem constants (match reference)
static constexpr int B = 32, T = 128, D = 128, U = 128, O = 10;
static constexpr float LAMB = 0.105f;
static constexpr float APERTURE = 1.0f / 3600.0f;   // 60^-2
static constexpr float LN_EPS = 1e-6f;

typedef float v2f __attribute__((ext_vector_type(2)));
typedef float v8f __attribute__((ext_vector_type(8)));

// ---------------------------------------------------------------------------
// Kernel 1: input projection GEMM with f32 WMMA (exact f32, matches reference).
//   XK[m, u] = sum_d X[m, d] * K[d, u] + bias[u],  m in [0, B*T)
// One wave per 16x16 output tile, V_WMMA_F32_16X16X4_F32 K-loop.
// ---------------------------------------------------------------------------
__global__ __launch_bounds__(32)
void xk_gemm_wmma(const float* __restrict__ X,
                  const float* __restrict__ K,
                  const float* __restrict__ bias,
                  float* __restrict__ XK) {
    const int row0 = blockIdx.x * 16;   // over M = B*T
    const int col0 = blockIdx.y * 16;   // over U
    const int lane = threadIdx.x & 31;
    const int m    = lane & 15;
    const int hi   = lane >> 4;

    v8f c = {};
    #pragma unroll 4
    for (int k0 = 0; k0 < D; k0 += 4) {
        const int ka = k0 + hi * 2;
        v2f a, b;
        a.x = X[(row0 + m) * D + ka];
        a.y = X[(row0 + m) * D + ka + 1];
        b.x = K[ka * U + col0 + m];
        b.y = K[(ka + 1) * U + col0 + m];
        c = __builtin_amdgcn_wmma_f32_16x16x4_f32(
                /*neg_a=*/false, a, /*neg_b=*/false, b,
                /*c_mod=*/(short)0, c, /*reuse_a=*/false, /*reuse_b=*/false);
    }

    const float bv = bias[col0 + m];
    #pragma unroll
    for (int r = 0; r < 8; ++r) {
        XK[(row0 + hi * 8 + r) * U + col0 + m] = c[r] + bv;
    }
}

// ---------------------------------------------------------------------------
// Kernel 2: sequential fast-weight scan. One workgroup per batch element.
//  - C (64KB) and R (64KB) both resident in LDS (128KB of 320KB/WGP;
//    2 workgroups/WGP still fit).
//  - 256 threads = 8 waves (2/SIMD): column j owned by threads (j, half=0/1),
//    each covering half the i-range -> 64-iteration serial chains + 2-way
//    LDS partial reduction. All C/R accesses conflict-free (lane = bank),
//    all h/diff reads are same-address broadcasts.
// ---------------------------------------------------------------------------
__global__ __launch_bounds__(256)
void fastweight_scan(const float* __restrict__ XK,   // (B*T, U), bias folded in
                     const float* __restrict__ R,    // (U, U) recurrent kernel
                     const float* __restrict__ gamma,
                     const float* __restrict__ beta,
                     const float* __restrict__ dw,   // (U, O)
                     const float* __restrict__ db,   // (O)
                     float* __restrict__ out) {      // (B, O)
    __shared__ float sC[U * U];        // 64 KB fast-weight matrix
    __shared__ float sR[U * U];        // 64 KB recurrent kernel (read-only)
    __shared__ float sh[U];            // h
    __shared__ float sdiff[U];         // h - hC
    __shared__ float sf[U];            // filtered
    __shared__ float spart[2 * U];     // 2-way partial reductions
    __shared__ float slg[O];           // logits

    const int b    = blockIdx.x;
    const int tid  = threadIdx.x;          // 0..255
    const int j    = tid & (U - 1);        // column 0..127
    const int half = tid >> 7;             // 0 or 1
    const int i0   = half * (U / 2);       // this thread's i-range start

    // Stage R into LDS once (coalesced b128 loads).
    {
        const float4* R4  = (const float4*)R;
        float4*       sR4 = (float4*)sR;
        #pragma unroll
        for (int idx = tid; idx < (U * U) / 4; idx += 256)
            sR4[idx] = R4[idx];
    }
    // C = I (each thread initializes its half-column), h0 = 0.
    for (int i = i0; i < i0 + U / 2; ++i)
        sC[i * U + j] = (i == j) ? 1.0f : 0.0f;
    if (half == 0) sh[j] = 0.0f;
    __syncthreads();

    const float gam = gamma[j];
    const float bet = beta[j];

    for (int t = 0; t < T; ++t) {
        // ---- pre[j] = XK[b,t,j] + sum_i h[i]*R[i][j]  (split over halves)
        float p = (half == 0) ? XK[(b * T + t) * U + j] : 0.0f;
        #pragma unroll 8
        for (int i = i0; i < i0 + U / 2; ++i)
            p = fmaf(sh[i], sR[i * U + j], p);
        spart[half * U + j] = p;
        if (t + 1 < T)  // warm next step's XK row (global_prefetch_b8)
            __builtin_prefetch(&XK[(b * T + t + 1) * U + j], 0, 0);
        __syncthreads();                       // partials ready; sh reads done
        if (half == 0)
            sh[j] = tanhf(spart[j] + spart[U + j]);
        __syncthreads();                       // new h visible

        // ---- hC[j] = sum_i h[i]*C[i][j] ; diff[j] = h[j] - hC[j]
        float a = 0.0f;
        #pragma unroll 8
        for (int i = i0; i < i0 + U / 2; ++i)
            a = fmaf(sh[i], sC[i * U + j], a);
        spart[half * U + j] = a;
        __syncthreads();
        if (half == 0)
            sdiff[j] = sh[j] - (spart[j] + spart[U + j]);
        __syncthreads();

        // ---- C[i][j] += LAMB*(diff[i]*h[j] - APERTURE*C[i][j]),
        //      filtered[j] = sum_i h[i]*Cnew[i][j]   (fused, split over halves)
        const float hj = sh[j];
        float f = 0.0f;
        #pragma unroll 4
        for (int i = i0; i < i0 + U / 2; ++i) {
            float c = sC[i * U + j];
            c = fmaf(LAMB, fmaf(sdiff[i], hj, -APERTURE * c), c);
            sC[i * U + j] = c;
            f = fmaf(sh[i], c, f);
        }
        spart[half * U + j] = f;
        __syncthreads();                       // partials ready; sh/sdiff reads done
        if (half == 0)
            sf[j] = spart[j] + spart[U + j];
        __syncthreads();                       // sf visible

        // ---- LayerNorm over the 128 filtered values (single fused pass:
        //      E[x], E[x^2]; broadcast LDS reads)
        float s1 = 0.0f, s2 = 0.0f;
        #pragma unroll 8
        for (int i = 0; i < U; ++i) {
            const float v = sf[i];
            s1 += v;
            s2 = fmaf(v, v, s2);
        }
        const float mu  = s1 * (1.0f / U);
        const float var = fmaf(s2, 1.0f / U, -mu * mu);
        if (half == 0)                          // safe: last sh reads were pre-barrier
            sh[j] = tanhf((sf[j] - mu) * rsqrtf(var + LN_EPS) * gam + bet);
        __syncthreads();                        // end of step: new h visible
    }

    // ---- logits + softmax (lanes 0..9)
    if (tid < O) {
        float lg = db[tid];
        #pragma unroll 8
        for (int i = 0; i < U; ++i)
            lg = fmaf(sh[i], dw[i * O + tid], lg);
        slg[tid] = lg;
    }
    __syncthreads();
    if (tid < O) {
        float mx = slg[0];
        #pragma unroll
        for (int o = 1; o < O; ++o) mx = fmaxf(mx, slg[o]);
        float ssum = 0.0f;
        #pragma unroll
        for (int o = 0; o < O; ++o) ssum += __expf(slg[o] - mx);
        out[b * O + tid] = __expf(slg[tid] - mx) / ssum;
    }
}

// ---------------------------------------------------------------------------
extern "C" void kernel_launch(void* const* d_in, const int* in_sizes, int n_in,
                              void* d_out, int out_size, void* d_ws, size_t ws_size,
                              hipStream_t stream) {
    const float* x     = (const float*)d_in[0];  // (B,T,D)
    const float* kern  = (const float*)d_in[1];  // (D,U)
    const float* R     = (const float*)d_in[2];  // (U,U)
    const float* bias  = (const float*)d_in[3];  // (U)
    const float* gamma = (const float*)d_in[4];  // (U)
    const float* beta  = (const float*)d_in[5];  // (U)
    const float* dw    = (const float*)d_in[6];  // (U,O)
    const float* db    = (const float*)d_in[7];  // (O)
    float* out = (float*)d_out;

    float* XK = (float*)d_ws;                    // (B*T, U) = 2 MB scratch

    dim3 g1((B * T) / 16, U / 16);               // 256 x 8 tiles
    xk_gemm_wmma<<<g1, 32, 0, stream>>>(x, kern, bias, XK);

    fastweight_scan<<<B, 256, 0, stream>>>(XK, R, gamma, beta, dw, db, out);
}